// STRMoE_23983097381216
// MI455X (gfx1250) — compile-verified
//
#include <hip/hip_runtime.h>
#include <math.h>

// ---------------- problem constants ----------------
constexpr int B_ = 16;
constexpr int S_ = 512;
constexpr int D_ = 768;
constexpr int E_ = 16;
constexpr int K_ = 4;

// ---------------- WMMA types (wave32, gfx1250) ----------------
typedef __attribute__((ext_vector_type(2))) float v2f;
typedef __attribute__((ext_vector_type(8))) float v8f;
typedef __attribute__((ext_vector_type(2))) int   ai2;   // async builtin arg types
typedef __attribute__((ext_vector_type(4))) int   ai4;

// ---------------- GEMM tiling ----------------
#define BM 128           // rows (s) per workgroup tile
#define BN 128           // cols (o) per workgroup tile
#define BK 16            // k (i) per stage
#define PA 20            // LDS A row stride (floats); keeps b64/b128 alignment
#define PBP 264          // LDS B pair-row stride (floats) = 2*BN + 8 ; %64==8

// Optional CDNA5 async memory->LDS path (compile-safe feature detection).
#if __has_builtin(__builtin_amdgcn_global_load_async_to_lds_b128) && \
    __has_builtin(__builtin_amdgcn_global_load_async_to_lds_b64)  && \
    __has_builtin(__builtin_amdgcn_s_wait_asynccnt)
#define USE_ASYNC_LDS 1
#define GLOBAL_AS __attribute__((address_space(1)))
#define LDS_AS    __attribute__((address_space(3)))
#endif

// =====================================================================
// Kernel 1: xm[b,d] = mean over s of x[b,s,d]
// =====================================================================
__global__ void moe_mean(const float* __restrict__ x, float* __restrict__ xm) {
    int idx = blockIdx.x * 256 + threadIdx.x;          // 0 .. B_*D_-1
    int b = idx / D_;
    int d = idx - b * D_;
    const float* p = x + (size_t)b * S_ * D_ + d;
    float a0 = 0.f, a1 = 0.f, a2 = 0.f, a3 = 0.f;
    for (int s = 0; s < S_; s += 4) {
        a0 += p[(size_t)(s + 0) * D_];
        a1 += p[(size_t)(s + 1) * D_];
        a2 += p[(size_t)(s + 2) * D_];
        a3 += p[(size_t)(s + 3) * D_];
    }
    xm[idx] = (a0 + a1 + a2 + a3) * (1.0f / (float)S_);
}

// =====================================================================
// Kernel 2: full gating path + loss. One block of 256 threads,
// thread (b,e) with tid = b*16+e.
// =====================================================================
__global__ void moe_gate(const float* __restrict__ xm,
                         const float* __restrict__ noise,
                         const float* __restrict__ w_gate,
                         const float* __restrict__ w_noise,
                         const float* __restrict__ sth,
                         float* __restrict__ gates_out,
                         float* __restrict__ loss_out) {
    __shared__ float s_noisy[B_ * E_];
    __shared__ float s_g[B_ * E_];
    __shared__ float s_prob[B_ * E_];
    __shared__ float s_thr_in[B_], s_thr_out[B_];
    __shared__ float s_load[E_], s_imp[E_];

    const int tid = threadIdx.x;
    const int b = tid >> 4;
    const int e = tid & 15;

    float c = 0.f, n = 0.f;
    const float* xr = xm + b * D_;
    for (int d = 0; d < D_; ++d) {
        float xv = xr[d];
        c += xv * w_gate[d * E_ + e];
        n += xv * w_noise[d * E_ + e];
    }
    float stddev = fmaxf(n, 0.f) + log1pf(expf(-fabsf(n))) + 0.01f;  // softplus+eps
    float nz = c + noise[tid] * stddev;
    s_noisy[tid] = nz;
    __syncthreads();

    float m = -INFINITY;
    for (int j = 0; j < E_; ++j) m = fmaxf(m, s_noisy[b * E_ + j]);
    float sum = 0.f;
    for (int j = 0; j < E_; ++j) sum += expf(s_noisy[b * E_ + j] - m);
    float g = expf(nz - m) / sum;
    float thr = 1.0f / (1.0f + expf(-sth[0]));
    g = fmaxf(g - thr, 0.f);
    s_g[tid] = g;
    gates_out[tid] = g;

    if (e == 0) {
        float a[E_];
        for (int j = 0; j < E_; ++j) a[j] = s_noisy[b * E_ + j];
        for (int i = 1; i < E_; ++i) {          // insertion sort, descending
            float v = a[i];
            int j = i - 1;
            while (j >= 0 && a[j] < v) { a[j + 1] = a[j]; --j; }
            a[j + 1] = v;
        }
        s_thr_in[b]  = a[K_];       // (K+1)-th largest
        s_thr_out[b] = a[K_ - 1];   // K-th largest
    }
    __syncthreads();

    float t_in = s_thr_in[b], t_out = s_thr_out[b];
    bool is_in = nz > t_in;
    float z = (c - (is_in ? t_in : t_out)) / stddev;
    float prob = 0.5f * erfcf(-z * 0.70710678118654752f);   // normcdf
    s_prob[tid] = prob;
    __syncthreads();

    if (tid < E_) {
        float L = 0.f, I = 0.f;
        for (int bb = 0; bb < B_; ++bb) {
            L += s_prob[bb * E_ + tid];
            I += s_g[bb * E_ + tid];
        }
        s_load[tid] = L;
        s_imp[tid] = I;
    }
    __syncthreads();

    if (tid == 0) {
        float loss = 0.f;
        for (int pass = 0; pass < 2; ++pass) {
            const float* v = (pass == 0) ? s_imp : s_load;
            float mean = 0.f;
            for (int j = 0; j < E_; ++j) mean += v[j];
            mean *= (1.0f / (float)E_);
            float var = 0.f;
            for (int j = 0; j < E_; ++j) { float d = v[j] - mean; var += d * d; }
            var *= (1.0f / (float)(E_ - 1));                 // ddof = 1
            loss += var / (mean * mean + 1e-10f);
        }
        loss_out[0] = loss * 0.01f;
    }
}

// =====================================================================
// Kernel 3: Wexp[b,o,i] = sum_e g[b,e] * W[e,o,i]; weight read exactly once.
// =====================================================================
__global__ void moe_combine_w(const float* __restrict__ g,
                              const float* __restrict__ w,
                              float* __restrict__ wexp) {
    __shared__ float sg[B_ * E_];
    const int tid = threadIdx.x;
    sg[tid] = g[tid];
    __syncthreads();
    const size_t idx = (size_t)blockIdx.x * 256 + tid;     // flat (o,i)
    float acc[B_];
    #pragma unroll
    for (int b = 0; b < B_; ++b) acc[b] = 0.f;
    #pragma unroll 4
    for (int e = 0; e < E_; ++e) {
        float wv = w[(size_t)e * D_ * D_ + idx];
        #pragma unroll
        for (int b = 0; b < B_; ++b) acc[b] += sg[b * E_ + e] * wv;
    }
    #pragma unroll
    for (int b = 0; b < B_; ++b) wexp[(size_t)b * D_ * D_ + idx] = acc[b];
}

__global__ void moe_combine_b(const float* __restrict__ g,
                              const float* __restrict__ bias,
                              float* __restrict__ bexp) {
    int idx = blockIdx.x * 256 + threadIdx.x;     // 0 .. B_*D_-1
    int b = idx / D_;
    int o = idx - b * D_;
    float acc = 0.f;
    #pragma unroll
    for (int e = 0; e < E_; ++e) acc += g[b * E_ + e] * bias[e * D_ + o];
    bexp[idx] = acc;
}

// =====================================================================
// Kernel 4: y[b] = x[b] (512x768) * Wexp[b]^T (768x768) + bexp[b]
// fp32 WMMA 16x16x4, 128x128 tile, 8 waves of 32x64, double-buffered LDS.
// B stored K-pair interleaved: sB[(k>>1)*PBP + n*2 + (k&1)] so one aligned
// ds_load_b64 yields a whole B fragment (no register re-packing).
// =====================================================================
__device__ __forceinline__ void gemm_load(const float* __restrict__ gA,
                                          const float* __restrict__ gB,
                                          int k0, float* sAbuf, float* sBbuf,
                                          int lr, int lc) {
    #pragma unroll
    for (int h = 0; h < 2; ++h) {
        const int r = lr + h * 64;                          // row within tile
        const float* pA = gA + (size_t)r * D_ + k0 + lc;
        const float* pB = gB + (size_t)r * D_ + k0 + lc;
        float* lA = sAbuf + r * PA + lc;
        float* lB0 = sBbuf + (lc >> 1) * PBP + r * 2;       // pair (lc,lc+1)
        float* lB1 = sBbuf + ((lc >> 1) + 1) * PBP + r * 2; // pair (lc+2,lc+3)
#ifdef USE_ASYNC_LDS
        // CDNA5 async memory->LDS (ASYNCcnt-tracked, no VGPR staging)
        __builtin_amdgcn_global_load_async_to_lds_b128(
            (GLOBAL_AS ai4*)pA, (LDS_AS ai4*)lA, 0, 0);
        __builtin_amdgcn_global_load_async_to_lds_b64(
            (GLOBAL_AS ai2*)pB, (LDS_AS ai2*)lB0, 0, 0);
        __builtin_amdgcn_global_load_async_to_lds_b64(
            (GLOBAL_AS ai2*)(pB + 2), (LDS_AS ai2*)lB1, 0, 0);
#else
        float4 va = *(const float4*)pA;
        *(float4*)lA = va;
        float4 vb = *(const float4*)pB;
        v2f t0; t0.x = vb.x; t0.y = vb.y;
        v2f t1; t1.x = vb.z; t1.y = vb.w;
        *(v2f*)lB0 = t0;
        *(v2f*)lB1 = t1;
#endif
    }
    if (k0 + BK < D_) {   // warm the tile after next (global_prefetch_b8)
        __builtin_prefetch(gA + (size_t)lr * D_ + k0 + BK, 0, 1);
        __builtin_prefetch(gB + (size_t)lr * D_ + k0 + BK, 0, 1);
    }
}

__device__ __forceinline__ void gemm_sync_stage() {
#ifdef USE_ASYNC_LDS
    __builtin_amdgcn_s_wait_asynccnt(0);   // our async LDS writes landed
#endif
    __syncthreads();
}

__device__ __forceinline__ void gemm_compute(const float* sAbuf,
                                             const float* sBbuf,
                                             v8f (&acc)[2][4],
                                             int lane, int wm, int wn) {
    const int l15 = lane & 15;
    const int kh = (lane >> 4) << 1;   // lanes 0-15 -> K 0,1 ; lanes 16-31 -> K 2,3
    #pragma unroll
    for (int kk = 0; kk < BK; kk += 4) {
        v2f af[2];
        #pragma unroll
        for (int mt = 0; mt < 2; ++mt) {
            const int row = wm * 32 + mt * 16 + l15;
            af[mt] = *(const v2f*)(sAbuf + row * PA + kk + kh);   // ds_load_b64
        }
        v2f bf[4];
        const int prow = (kk + kh) >> 1;                // K-pair row (kk+kh even)
        #pragma unroll
        for (int nt = 0; nt < 4; ++nt) {
            const int col = wn * 64 + nt * 16 + l15;
            bf[nt] = *(const v2f*)(sBbuf + prow * PBP + col * 2); // ds_load_b64
        }
        #pragma unroll
        for (int mt = 0; mt < 2; ++mt)
            #pragma unroll
            for (int nt = 0; nt < 4; ++nt)
                acc[mt][nt] = __builtin_amdgcn_wmma_f32_16x16x4_f32(
                    false, af[mt], false, bf[nt], (short)0, acc[mt][nt],
                    false, false);
    }
}

__global__ __launch_bounds__(256)
void moe_gemm(const float* __restrict__ x, const float* __restrict__ wexp,
              const float* __restrict__ bexp, float* __restrict__ y) {
    __shared__ float sA[2][BM * PA];
    __shared__ float sB[2][(BK / 2) * PBP];

    const int b  = blockIdx.z;
    const int s0 = blockIdx.y * BM;
    const int o0 = blockIdx.x * BN;
    const int tid = threadIdx.x;
    const int lane = tid & 31;
    const int wave = tid >> 5;
    const int wm = wave & 3;     // 4 wave rows  (32 s-rows each)
    const int wn = wave >> 2;    // 2 wave cols  (64 o-cols each)

    const float* gA = x    + (size_t)b * S_ * D_ + (size_t)s0 * D_;
    const float* gB = wexp + (size_t)b * D_ * D_ + (size_t)o0 * D_;

    const int lr = tid >> 2;          // 0..63 : tile row for loads
    const int lc = (tid & 3) * 4;     // 0,4,8,12 : k column for loads

    v8f acc[2][4];
    const v8f vzero = {0.f, 0.f, 0.f, 0.f, 0.f, 0.f, 0.f, 0.f};
    #pragma unroll
    for (int mt = 0; mt < 2; ++mt)
        #pragma unroll
        for (int nt = 0; nt < 4; ++nt) acc[mt][nt] = vzero;

    gemm_load(gA, gB, 0, sA[0], sB[0], lr, lc);
    gemm_sync_stage();

    int cur = 0;
    for (int k0 = BK; k0 < D_; k0 += BK) {
        gemm_load(gA, gB, k0, sA[cur ^ 1], sB[cur ^ 1], lr, lc);
        gemm_compute(sA[cur], sB[cur], acc, lane, wm, wn);
        gemm_sync_stage();
        cur ^= 1;
    }
    gemm_compute(sA[cur], sB[cur], acc, lane, wm, wn);

    // epilogue: C/D layout -> VGPR r holds M=r (lanes 0-15) / M=8+r (lanes 16-31)
    const int l15 = lane & 15;
    const int mh = (lane >> 4) * 8;
    float* yb = y + (size_t)b * S_ * D_;
    #pragma unroll
    for (int mt = 0; mt < 2; ++mt) {
        const int srow = s0 + wm * 32 + mt * 16 + mh;
        #pragma unroll
        for (int nt = 0; nt < 4; ++nt) {
            const int o = o0 + wn * 64 + nt * 16 + l15;
            const float bv = bexp[b * D_ + o];
            #pragma unroll
            for (int r = 0; r < 8; ++r)
                yb[(size_t)(srow + r) * D_ + o] = acc[mt][nt][r] + bv;
        }
    }
}

// =====================================================================
// Host launcher
// =====================================================================
extern "C" void kernel_launch(void* const* d_in, const int* in_sizes, int n_in,
                              void* d_out, int out_size, void* d_ws, size_t ws_size,
                              hipStream_t stream) {
    const float* x       = (const float*)d_in[0];
    const float* noise   = (const float*)d_in[1];
    const float* w_gate  = (const float*)d_in[2];
    const float* w_noise = (const float*)d_in[3];
    const float* sth     = (const float*)d_in[4];
    const float* weight  = (const float*)d_in[5];
    const float* bias    = (const float*)d_in[6];

    float* y    = (float*)d_out;
    float* loss = y + (size_t)B_ * S_ * D_;

    float* ws    = (float*)d_ws;
    float* xm    = ws;                                   // 16*768
    float* gates = ws + B_ * D_;                         // 16*16
    float* bexp  = gates + B_ * E_;                      // 16*768
    float* wexp  = bexp + B_ * D_;                       // 16*768*768

    moe_mean<<<(B_ * D_) / 256, 256, 0, stream>>>(x, xm);
    moe_gate<<<1, 256, 0, stream>>>(xm, noise, w_gate, w_noise, sth, gates, loss);
    moe_combine_w<<<(D_ * D_) / 256, 256, 0, stream>>>(gates, weight, wexp);
    moe_combine_b<<<(B_ * D_) / 256, 256, 0, stream>>>(gates, bias, bexp);

    dim3 grid(D_ / BN, S_ / BM, B_);   // (6, 4, 16)
    moe_gemm<<<grid, 256, 0, stream>>>(x, wexp, bexp, y);
}